// SpikingActivation_43267500540509
// MI455X (gfx1250) — compile-verified
//
#include <hip/hip_runtime.h>

// SpikingActivation forward: v += relu(x)*dt; n = floor(v); v -= n; out = n/dt
// Parallel-prefix reformulation per 16-step x 16-channel tile:
//   S = L(16x16 lower-tri, entries=dt) @ relu(X)   via 4x V_WMMA_F32_16X16X4_F32
//   n_t = floor(v + S_t) - floor(v + S_{t-1}),  carry v' = frac(v + S_15).

typedef __attribute__((ext_vector_type(2))) float v2f;
typedef __attribute__((ext_vector_type(8))) float v8f;

#define SPK_DT     0.001f
#define SPK_INVDT  1000.0f

constexpr int Bb = 32, Tt = 2048, Uu = 512;
constexpr int CH_PER_WAVE   = 16;                       // channels per wave
constexpr int WAVES_PER_BLK = 8;                        // 256 threads = 8 wave32
constexpr int TOTAL_WAVES   = Bb * (Uu / CH_PER_WAVE);  // 1024 waves
constexpr int NTILE         = Tt / 16;                  // 128 tiles
constexpr int PFD           = 2;                        // prefetch depth (tiles)

// single-instruction relu: v_med3_f32(x, 0, +inf) (no canonicalize pair)
__device__ __forceinline__ float relu1(float x) {
  return __builtin_amdgcn_fmed3f(x, 0.0f, __builtin_inff());
}

// identity cross-half exchange: lane L <- lane (L ^ 16), via v_permlanex16_b32
__device__ __forceinline__ float permx16(float s) {
  return __int_as_float(__builtin_amdgcn_permlanex16(
      __float_as_int(s), __float_as_int(s), 0x76543210u, 0xfedcba98u,
      false, false));
}

__global__ __launch_bounds__(WAVES_PER_BLK * 32)
void SpikingActivation_wmma_kernel(const float* __restrict__ x,
                                   const float* __restrict__ v0,
                                   float* __restrict__ out) {
  const int lane = threadIdx.x & 31;
  const int wid  = blockIdx.x * WAVES_PER_BLK + (threadIdx.x >> 5); // 0..1023
  const int b    = wid >> 5;
  const int u0   = (wid & 31) * CH_PER_WAVE;

  const int n  = lane & 15;        // channel (A row M / B col N / D col N)
  const int kh = (lane >> 4) * 2;  // K offset of this lane-half (f32 A/B layout)

  // Lower-triangular constant A with dt folded in; A[ko] = cols K=4ko..4ko+3.
  v2f A[4];
#pragma unroll
  for (int ko = 0; ko < 4; ++ko) {
    A[ko].x = ((4 * ko + kh + 0) <= n) ? SPK_DT : 0.0f;
    A[ko].y = ((4 * ko + kh + 1) <= n) ? SPK_DT : 0.0f;
  }

  // voltage carry, replicated in both lane halves; v in [0,1) => floor(v)==0
  float v = v0[b * Uu + u0 + n];

  const float* xp = x   + (size_t)b * Tt * Uu + u0 + n;
  float*       op = out + (size_t)b * Tt * Uu + u0 + n;
  const int mrow = (lane >> 4) * 8;   // D rows handled by this lane-half

  // ---- depth-2 software pipeline: tile t+2 loads issued while computing t ----
  float raw[PFD][8];
#pragma unroll
  for (int p = 0; p < PFD; ++p) {
    const float* tp = xp + (size_t)p * 16 * Uu;
#pragma unroll
    for (int j = 0; j < 8; ++j) {
      const int kk = 4 * (j >> 1) + kh + (j & 1);
      raw[p][j] = __builtin_nontemporal_load(&tp[(size_t)kk * Uu]);
    }
  }

#pragma unroll 2   // makes t%PFD compile-time const -> buffers stay in VGPRs
  for (int t = 0; t < NTILE; ++t) {
    // consume buffer: relu into B-matrix layout (lane = col N, K = 4ko+kh+{0,1})
    v2f Bm[4];
#pragma unroll
    for (int ko = 0; ko < 4; ++ko) {
      Bm[ko].x = relu1(raw[t % PFD][2 * ko + 0]);
      Bm[ko].y = relu1(raw[t % PFD][2 * ko + 1]);
    }

    // prefetch tile t+PFD into the freed buffer (clamped; tail re-loads last
    // tile instead of branching, keeps the loop body uniform)
    {
      const int tn = (t + PFD < NTILE) ? (t + PFD) : (NTILE - 1);
      const float* tp = xp + (size_t)tn * 16 * Uu;
#pragma unroll
      for (int j = 0; j < 8; ++j) {
        const int kk = 4 * (j >> 1) + kh + (j & 1);
        raw[t % PFD][j] = __builtin_nontemporal_load(&tp[(size_t)kk * Uu]);
      }
    }

    // inclusive prefix sums (dt units): S = sum_ko A[ko] x Bm[ko]
    v8f S = {};
#pragma unroll
    for (int ko = 0; ko < 4; ++ko)
      S = __builtin_amdgcn_wmma_f32_16x16x4_f32(false, A[ko], false, Bm[ko],
                                                (short)0, S, false, false);

    // S[j] -> timestep m = mrow + j, channel n
    float w[8], f[8];
#pragma unroll
    for (int j = 0; j < 8; ++j) {
      w[j] = v + S[j];
      f[j] = floorf(w[j]);
    }

    // upper half (m=8..15) needs floor at m=7 from the lower half
    const float f7o  = permx16(f[7]);
    float       prev = (lane < 16) ? 0.0f : f7o;   // floor(v) == 0

    float* tob = op + (size_t)t * 16 * Uu;
#pragma unroll
    for (int j = 0; j < 8; ++j) {
      __builtin_nontemporal_store((f[j] - prev) * SPK_INVDT,
                                  &tob[(size_t)(mrow + j) * Uu]);
      prev = f[j];
    }

    // carry: v' = frac(v + S[15]); lives in upper half at j=7.  Upper lanes
    // already own it; lower lanes fetch it via permlanex16.
    const float c  = w[7] - f[7];
    const float co = permx16(c);
    v = (lane < 16) ? co : c;
  }
}

extern "C" void kernel_launch(void* const* d_in, const int* in_sizes, int n_in,
                              void* d_out, int out_size, void* d_ws, size_t ws_size,
                              hipStream_t stream) {
  (void)in_sizes; (void)n_in; (void)out_size; (void)d_ws; (void)ws_size;
  const float* x  = (const float*)d_in[0];
  const float* v0 = (const float*)d_in[1];
  float* out      = (float*)d_out;

  dim3 grid(TOTAL_WAVES / WAVES_PER_BLK);   // 128 blocks
  dim3 block(WAVES_PER_BLK * 32);           // 256 threads = 8 wave32
  hipLaunchKernelGGL(SpikingActivation_wmma_kernel, grid, block, 0, stream,
                     x, v0, out);
}